// Transport_65309272703473
// MI455X (gfx1250) — compile-verified
//
#include <hip/hip_runtime.h>

// ---------------------------------------------------------------------------
// Transport kernel for MI455X (gfx1250).
// B=4, N=16384, P=32, K=16 (fixed by the reference).
// Output [B,N,P,K,2,2] f32 = 512 MiB -> store-bandwidth bound (~23us @ 23.3TB/s).
// Strategy: trig-free harmonic recurrence + LDS-staged output drained with the
// CDNA5 async LDS->global store engine (ASYNCcnt) for 512B/wave coalescing.
// ---------------------------------------------------------------------------

#define AS1 __attribute__((address_space(1)))
#define AS3 __attribute__((address_space(3)))

typedef int v4i_cdna5 __attribute__((ext_vector_type(4)));

#define BDIM   256
#define WAVES  8        // (b,n) sites per block (one per wave)
#define KH     16       // harmonics
#define PN     32       // neighbors per site == wave32 lanes
#define NLOG2  14       // N = 16384

__device__ __forceinline__ void async_store_b128(float* gptr, float* lptr) {
    // global_store_async_from_lds_b128: 16 bytes per lane, LDS -> global.
    // Builtin signature (from clang diagnostic): (v4i as1*, v4i as3*, Ii, Ii).
    __builtin_amdgcn_global_store_async_from_lds_b128(
        (AS1 v4i_cdna5*)gptr, (AS3 v4i_cdna5*)lptr, /*offset=*/0, /*cpol=*/0);
}

__global__ __launch_bounds__(BDIM)
void Transport_65309272703473_kernel(const float* __restrict__ frames,
                                     const int*   __restrict__ indices,
                                     float*       __restrict__ out)
{
    // 256 threads * 64 floats = 64 KB: exact image of this block's output slab.
    __shared__ float smem[BDIM * 4 * KH];

    const int tid  = threadIdx.x;
    const int lane = tid & (PN - 1);          // p
    const int wv   = tid >> 5;                // which (b,n) in block
    const int bn   = blockIdx.x * WAVES + wv; // flat b*N + n
    const int b    = bn >> NLOG2;

    // neighbor index (flat [B,N,P])
    const int nb = indices[(size_t)bn * PN + lane];

    const float* F = frames + (size_t)bn * 9;                      // center frame
    const float* G = frames + ((size_t)((b << NLOG2) + nb)) * 9;   // gathered frame

    // center: f0 = column 0 (tangent), n = column 2 (normal)
    const float f0x = F[0], f0y = F[3], f0z = F[6];
    const float n0  = F[2], n1  = F[5], n2  = F[8];

    // neighbor: all 9 (cols 0,1 for R2 projection, col 2 = normal)
    const float g00 = G[0], g01 = G[1], m0 = G[2];
    const float g10 = G[3], g11 = G[4], m1 = G[5];
    const float g20 = G[6], g21 = G[7], m2 = G[8];

    // q = n x m ; c = n . m
    const float q0 = n1 * m2 - n2 * m1;
    const float q1 = n2 * m0 - n0 * m2;
    const float q2 = n0 * m1 - n1 * m0;
    const float c  = n0 * m0 + n1 * m1 + n2 * m2;

    const float mask = (c + 0.9f > 0.0f) ? 1.0f : 0.0f;   // max(sign(c+0.9),0)
    const float c1   = 1.0f / fmaxf(1.0f + c, 1e-4f);

    // Rodrigues applied to f0 only (only column 0 of "transported" is needed):
    // Q x = x cross q  =>  t0 = f0 + (f0 x q) + c1 * ((f0 x q) x q)
    const float vx = f0y * q2 - f0z * q1;
    const float vy = f0z * q0 - f0x * q2;
    const float vz = f0x * q1 - f0y * q0;
    const float wx = vy * q2 - vz * q1;
    const float wy = vz * q0 - vx * q2;
    const float wz = vx * q1 - vy * q0;
    const float t0x = f0x + vx + c1 * wx;
    const float t0y = f0y + vy + c1 * wy;
    const float t0z = f0z + vz + c1 * wz;

    // R2[0,0] = Gcol0 . t0  (y), R2[1,0] = Gcol1 . t0  (x)
    const float yv = g00 * t0x + g10 * t0y + g20 * t0z;
    const float xv = g01 * t0x + g11 * t0y + g21 * t0z;

    // theta = -atan2(y, x)  =>  cos = x/r, sin = -y/r  (atan2(0,0)=0 case: c=1,s=0)
    const float r2 = xv * xv + yv * yv;
    float cth, sth;
    if (r2 > 0.0f) {
        const float ri = rsqrtf(r2);
        cth =  xv * ri;
        sth = -yv * ri;
    } else {
        cth = 1.0f;
        sth = 0.0f;
    }

    // harmonics cos(k*theta), sin(k*theta) via exact rotation recurrence;
    // stage [c,-s,s,c] quads into LDS mirroring the global layout.
    float* sp = &smem[tid * (4 * KH)];   // 256B per thread, 16B aligned
    float cj = cth, sj = sth;
#pragma unroll
    for (int j = 0; j < KH; ++j) {
        const float mc = mask * cj;
        const float ms = mask * sj;
        float4 v4 = make_float4(mc, -ms, ms, mc);
        *(float4*)(sp + 4 * j) = v4;          // ds_store_b128
        const float cn = cj * cth - sj * sth;
        const float sn = sj * cth + cj * sth;
        cj = cn; sj = sn;
    }

    __syncthreads();   // LDS image complete (waits DScnt + barrier)

    // Drain 64KB LDS slab with async LDS->global b128 copies:
    // chunk c = j*256 + tid -> consecutive lanes copy consecutive 16B chunks
    // => 512B fully coalesced per wave-instruction, tracked by ASYNCcnt.
    float* gbase = out + (size_t)blockIdx.x * (BDIM * 4 * KH);
#pragma unroll
    for (int j = 0; j < KH; ++j) {
        const int cidx = j * BDIM + tid;
        async_store_b128(gbase + (size_t)cidx * 4, &smem[cidx * 4]);
    }
    // S_ENDPGM performs an implicit wait-idle (covers ASYNCcnt) before the
    // workgroup's LDS is released.
}

extern "C" void kernel_launch(void* const* d_in, const int* in_sizes, int n_in,
                              void* d_out, int out_size, void* d_ws, size_t ws_size,
                              hipStream_t stream) {
    const float* frames  = (const float*)d_in[0];   // [B,N,3,3] f32
    const int*   indices = (const int*)d_in[1];     // [B,N,P]   i32
    float*       out     = (float*)d_out;           // [B,N,P,K,2,2] f32

    const int BN     = in_sizes[0] / 9;   // B*N = 65536
    const int blocks = BN / WAVES;        // 8192
    Transport_65309272703473_kernel<<<blocks, BDIM, 0, stream>>>(frames, indices, out);
}